// AEVComputer_35768487641377
// MI455X (gfx1250) — compile-verified
//
#include <hip/hip_runtime.h>

typedef __attribute__((ext_vector_type(2))) float v2f;
typedef __attribute__((ext_vector_type(8))) float v8f;

#define RCR 5.2f
#define RCA 3.5f
#define PI_F 3.14159265358979f

// packed upper-triangular species-pair index, symmetric (4 species -> 10 pairs)
__device__ __constant__ int c_triu[16] = {0,1,2,3, 1,4,5,6, 2,5,7,8, 3,6,8,9};
// cos/sin of SHF_Z = pi/16*(2t+1), t=0..7
__device__ __constant__ float c_cz[8] = { 0.98078528f, 0.83146961f, 0.55557023f, 0.19509032f,
                                         -0.19509032f,-0.55557023f,-0.83146961f,-0.98078528f};
__device__ __constant__ float c_sz[8] = { 0.19509032f, 0.55557023f, 0.83146961f, 0.98078528f,
                                          0.98078528f, 0.83146961f, 0.55557023f, 0.19509032f};

__global__ __launch_bounds__(32)
void aev_kernel(const int* __restrict__ species,
                const float* __restrict__ coords,
                float* __restrict__ out)
{
  const int lane = threadIdx.x;        // 0..31, one wave32 per block
  const int b    = blockIdx.x;         // center index: 0..511
  const int n    = b >> 6;
  const int i    = b & 63;
  const bool hi  = lane >= 16;
  const int  l   = lane & 15;

  __shared__ float s_cx[64], s_cy[64], s_czd[64];
  __shared__ int   s_sp[64];
  __shared__ float s_trm[64][16];                  // radial exp terms per j
  __shared__ float s_wR[64];                       // 0.25*fc per j
  __shared__ int   s_bR[64];                       // species bin per j
  __shared__ float s_nfc[64], s_nd[64];            // neighbor list (angular)
  __shared__ float s_nux[64], s_nuy[64], s_nuz[64];
  __shared__ int   s_nbin[64];
  __shared__ float s_ta[32][32];                   // angular terms per pair chunk
  __shared__ float s_wA[32];
  __shared__ int   s_bA[32];

  for (int jj = lane; jj < 64; jj += 32) {
    const float* cj = coords + (size_t)(n*64 + jj)*3;
    s_cx[jj] = cj[0]; s_cy[jj] = cj[1]; s_czd[jj] = cj[2];
    s_sp[jj] = species[n*64 + jj];
  }
  __syncthreads();

  const float xi = s_cx[i], yi = s_cy[i], zi = s_czd[i];

  // ---------------- radial terms ----------------
  for (int jj = lane; jj < 64; jj += 32) {
    float dx = s_cx[jj]-xi, dy = s_cy[jj]-yi, dz = s_czd[jj]-zi;
    float d  = sqrtf(dx*dx + dy*dy + dz*dz);
    bool valid = (jj != i) && (d <= RCR);
    float fc = valid ? (0.5f*__cosf(PI_F*d*(1.0f/RCR)) + 0.5f) : 0.f;
    s_wR[jj] = 0.25f*fc;
    s_bR[jj] = s_sp[jj];
    #pragma unroll
    for (int r = 0; r < 16; ++r) {
      float dd = d - (0.9f + 0.26875f*(float)r);   // SHF_R
      s_trm[jj][r] = __expf(-16.f*dd*dd);          // ETA_R
    }
  }
  __syncthreads();

  // D[s_bin(16 pad), shift(16)] += A[s_bin, j]*B[j, shift], K=4 per WMMA.
  // A layout (f32 16x4): lane<16 -> M=l, {x,y}={K0,K1}; lane>=16 -> M=l, {x,y}={K2,K3}.
  // Load w/bin into locals first so the one-hot select lowers to v_cndmask,
  // not exec-mask-predicated ds_loads.
  v8f accR = {};
  #pragma unroll
  for (int c = 0; c < 16; ++c) {
    int q0 = 4*c + (hi ? 2 : 0);
    int q1 = q0 + 1;
    float w0 = s_wR[q0], w1 = s_wR[q1];
    int   b0 = s_bR[q0], b1 = s_bR[q1];
    v2f A, B;
    A.x = (b0 == l) ? w0 : 0.f;
    A.y = (b1 == l) ? w1 : 0.f;
    B.x = s_trm[q0][l];
    B.y = s_trm[q1][l];
    accR = __builtin_amdgcn_wmma_f32_16x16x4_f32(false, A, false, B, (short)0, accR, false, false);
  }

  // ---------------- angular neighbor list ----------------
  int mbase = 0;
  for (int it = 0; it < 2; ++it) {
    int jj = it*32 + lane;
    float dx = s_cx[jj]-xi, dy = s_cy[jj]-yi, dz = s_czd[jj]-zi;
    float d  = sqrtf(dx*dx + dy*dy + dz*dz);
    bool valid = (jj != i) && (d <= RCA);
    unsigned mask = (unsigned)__ballot(valid);     // wave32: low 32 bits
    int pos = mbase + __popc(mask & ((1u << lane) - 1u));
    if (valid) {
      float inv = 1.f/d;
      s_nfc[pos] = 0.5f*__cosf(PI_F*d*(1.0f/RCA)) + 0.5f;
      s_nd[pos]  = d;
      s_nux[pos] = dx*inv; s_nuy[pos] = dy*inv; s_nuz[pos] = dz*inv;
      s_nbin[pos] = s_sp[jj];
    }
    mbase += __popc(mask);
  }
  const int M = mbase;
  const int P = (M*(M-1)) >> 1;                    // unordered neighbor pairs
  __syncthreads();

  v8f acc0 = {}, acc1 = {};
  for (int qb = 0; qb < P; qb += 32) {             // uniform loop bound
    int q = qb + lane;
    float w = 0.f; int bin = 0;
    if (q < P) {
      // unrank q -> (j<k)
      int j = 0, rem = q;
      while (rem >= M - 1 - j) { rem -= M - 1 - j; ++j; }
      int k = j + 1 + rem;
      float ca = 0.95f*(s_nux[j]*s_nux[k] + s_nuy[j]*s_nuy[k] + s_nuz[j]*s_nuz[k]);
      float sa = sqrtf(fmaxf(0.f, 1.f - ca*ca));   // sin(arccos(ca)), ang in [0,pi]
      float davg = 0.5f*(s_nd[j] + s_nd[k]);
      w   = 2.f*s_nfc[j]*s_nfc[k];
      bin = c_triu[s_nbin[j]*4 + s_nbin[k]];
      float f1[8];
      #pragma unroll
      for (int tz = 0; tz < 8; ++tz) {
        // cos(ang - shf_z) = ca*cos(shf) + sa*sin(shf); f1 = ((1+.)/2)^32
        float ct  = 0.5f*(1.f + ca*c_cz[tz] + sa*c_sz[tz]);
        float p2 = ct*ct, p4 = p2*p2, p8 = p4*p4, p16 = p8*p8;
        f1[tz] = p16*p16;
      }
      #pragma unroll
      for (int ta = 0; ta < 4; ++ta) {             // contiguous per-ta stores -> b128 merge
        float dd = davg - (0.9f + 0.65f*(float)ta); // SHF_A
        float f2 = __expf(-8.f*dd*dd);              // ETA_A
        #pragma unroll
        for (int tz = 0; tz < 8; ++tz) s_ta[lane][ta*8 + tz] = f2*f1[tz];
      }
    } else {
      #pragma unroll
      for (int t = 0; t < 32; ++t) s_ta[lane][t] = 0.f;
    }
    s_wA[lane] = w;
    s_bA[lane] = bin;
    __syncthreads();

    // D[pair_bin(16 pad), term(2x16)] over K=4 pairs per WMMA, 8 chunks of 32 pairs
    #pragma unroll
    for (int c = 0; c < 8; ++c) {
      int q0 = 4*c + (hi ? 2 : 0);
      int q1 = q0 + 1;
      float w0 = s_wA[q0], w1 = s_wA[q1];
      int   b0 = s_bA[q0], b1 = s_bA[q1];
      v2f A, B0, B1;
      A.x = (b0 == l) ? w0 : 0.f;
      A.y = (b1 == l) ? w1 : 0.f;
      B0.x = s_ta[q0][l];      B0.y = s_ta[q1][l];
      B1.x = s_ta[q0][l+16];   B1.y = s_ta[q1][l+16];
      acc0 = __builtin_amdgcn_wmma_f32_16x16x4_f32(false, A, false, B0, (short)0, acc0, false, false);
      acc1 = __builtin_amdgcn_wmma_f32_16x16x4_f32(false, A, false, B1, (short)0, acc1, false, false);
    }
    __syncthreads();
  }

  // ---------------- write out ----------------
  // features: [0,64) radial = s*16+r ; [64,384) angular = 64 + p*32 + t
  float* o = out + (size_t)b*384;
  if (!hi) {
    #pragma unroll
    for (int rr = 0; rr < 4; ++rr) o[rr*16 + l] = accR[rr];   // D[M=rr][N=l]
  }
  #pragma unroll
  for (int rr = 0; rr < 8; ++rr) {
    int p = rr + (hi ? 8 : 0);                                 // D row for this lane half
    if (p < 10) {
      o[64 + p*32 + l]      = acc0[rr];
      o[64 + p*32 + 16 + l] = acc1[rr];
    }
  }
}

extern "C" void kernel_launch(void* const* d_in, const int* in_sizes, int n_in,
                              void* d_out, int out_size, void* d_ws, size_t ws_size,
                              hipStream_t stream) {
  const int*   species = (const int*)d_in[0];      // (8,64) int32
  const float* coords  = (const float*)d_in[1];    // (8,64,3) f32
  float*       out     = (float*)d_out;            // (8,64,384) f32
  int centers = in_sizes[0];                       // N*A = 512
  aev_kernel<<<dim3(centers), dim3(32), 0, stream>>>(species, coords, out);
}